// PostProcessor_77257871720852
// MI455X (gfx1250) — compile-verified
//
#include <hip/hip_runtime.h>
#include <hip/hip_bf16.h>
#include <math.h>

typedef float v2f __attribute__((ext_vector_type(2)));
typedef float v8f __attribute__((ext_vector_type(8)));

#define NN   1024
#define NC   81
#define NCF  80
#define DET  100
#define SCORE_TH 0.05f
#define NMS_TH   0.5f
#define XFORM_CLIP 4.135166556742356f   // log(1000/16)
#define IMGW 1333.0f
#define IMGH 800.0f

// workspace layout (float-element offsets)
#define OFF_PROBS 0                       // 1024*81
#define OFF_SS    (NN*NC)                 // sorted scores  [80][1024]
#define OFF_SA    (OFF_SS + NCF*NN)       // sorted areas   [80][1024]
#define OFF_SB    (OFF_SA + NCF*NN)       // sorted boxes   [80][1024][4]
#define OFF_RANK  (OFF_SB + NCF*NN*4)     // ranked scores  [80*1024]
#define OFF_MASK  (OFF_RANK + NCF*NN)     // u32 bitmask    [80][1024][32]

// ---------------------------------------------------------------- kernel 1
__global__ void k_softmax(const float* __restrict__ logits, float* __restrict__ ws) {
  int r = blockIdx.x * blockDim.x + threadIdx.x;      // 0..1023
  const float* row = logits + r * NC;
  __builtin_prefetch(row, 0, 0);                      // global_prefetch_b8
  float m = -3.4e38f;
  for (int c = 0; c < NC; ++c) m = fmaxf(m, row[c]);
  float s = 0.f;
  float* pr = ws + OFF_PROBS + r * NC;
  for (int c = 0; c < NC; ++c) { float e = __expf(row[c] - m); pr[c] = e; s += e; }
  float inv = 1.0f / s;
  for (int c = 0; c < NC; ++c) pr[c] *= inv;
}

// ---------------------------------------------------------------- kernel 2
// one block per foreground class: bitonic-sort 1024 scores (desc), then
// decode+clip boxes in sorted order, emit scores/areas/boxes to ws.
__global__ void k_sort_decode(const float* __restrict__ rel,
                              const float* __restrict__ props,
                              float* __restrict__ ws) {
  __shared__ float ss[NN];
  __shared__ int   si[NN];
  const int c   = blockIdx.x;            // fg class -> real class c+1
  const int tid = threadIdx.x;           // 256 threads
  const float* probs = ws + OFF_PROBS;
  for (int i = tid; i < NN; i += 256) { ss[i] = probs[i * NC + (c + 1)]; si[i] = i; }
  __syncthreads();
  for (int k = 2; k <= NN; k <<= 1) {
    for (int j = k >> 1; j > 0; j >>= 1) {
      for (int i = tid; i < NN; i += 256) {
        int ixj = i ^ j;
        if (ixj > i) {
          float a = ss[i], b = ss[ixj];
          bool desc = ((i & k) == 0);
          bool sw = desc ? (a < b) : (a > b);
          if (sw) { ss[i] = b; ss[ixj] = a; int t = si[i]; si[i] = si[ixj]; si[ixj] = t; }
        }
      }
      __syncthreads();
    }
  }
  for (int slot = tid; slot < NN; slot += 256) {
    int o = si[slot];
    float p0 = props[o*4+0], p1 = props[o*4+1], p2 = props[o*4+2], p3 = props[o*4+3];
    float w = p2 - p0 + 1.0f, h = p3 - p1 + 1.0f;
    float cx = p0 + 0.5f*w,  cy = p1 + 0.5f*h;
    const float* r4 = rel + (size_t)o*(4*NC) + 4*(c+1);
    float dx = r4[0]*0.1f, dy = r4[1]*0.1f;
    float dw = fminf(r4[2]*0.2f, XFORM_CLIP), dh = fminf(r4[3]*0.2f, XFORM_CLIP);
    float pcx = dx*w + cx, pcy = dy*h + cy;
    float pw = __expf(dw)*w, ph = __expf(dh)*h;
    float x1 = pcx - 0.5f*pw,        y1 = pcy - 0.5f*ph;
    float x2 = pcx + 0.5f*pw - 1.0f, y2 = pcy + 0.5f*ph - 1.0f;
    x1 = fminf(fmaxf(x1, 0.f), IMGW-1.f); x2 = fminf(fmaxf(x2, 0.f), IMGW-1.f);
    y1 = fminf(fmaxf(y1, 0.f), IMGH-1.f); y2 = fminf(fmaxf(y2, 0.f), IMGH-1.f);
    int g = c*NN + slot;
    ws[OFF_SS + g] = ss[slot];
    ws[OFF_SA + g] = (x2-x1+1.f)*(y2-y1+1.f);
    float* b4 = ws + OFF_SB + (size_t)g*4;
    b4[0]=x1; b4[1]=y1; b4[2]=x2; b4[3]=y2;
  }
}

// ---------------------------------------------------------------- kernel 3
// suppression bitmask. grid = (64 row-tiles, 80 classes), 1 wave per block.
// union term area_i + area_j via V_WMMA_F32_16X16X4_F32 outer-sum:
//   A row m = (area_row[m], 1, 0, 0), B col n = (1, area_col[n], 0, 0)^T
__global__ void k_mask(float* __restrict__ ws) {
  __shared__ float4   rbox[16];
  __shared__ unsigned rw[16][32];
  const int c  = blockIdx.y;
  const int rowbase = blockIdx.x * 16;
  const int l   = threadIdx.x;          // 0..31, full wave
  const int l16 = l & 15;
  const int hi  = l >> 4;
  const float*  sA = ws + OFF_SA + c*NN;
  const float4* sB = (const float4*)(ws + OFF_SB) + (size_t)c*NN;
  unsigned* mask = (unsigned*)(ws + OFF_MASK) + (size_t)c*NN*32;

  for (int i = l; i < 16*32; i += 32) ((unsigned*)rw)[i] = 0u;
  if (l < 16) rbox[l] = sB[rowbase + l];
  __syncthreads();

  v2f a;                                 // A 16x4: lanes0-15 = K0,K1
  a.x = (hi == 0) ? sA[rowbase + l16] : 0.0f;
  a.y = (hi == 0) ? 1.0f : 0.0f;
  v8f cz = {};

  for (int ct = 0; ct < 64; ++ct) {
    int colbase = ct * 16;
    int col = colbase + l16;
    float4 cb = sB[col];
    v2f b;                               // B 4x16: lanes0-15 = K0,K1 rows
    b.x = (hi == 0) ? 1.0f : 0.0f;
    b.y = (hi == 0) ? sA[col] : 0.0f;
    v8f d = __builtin_amdgcn_wmma_f32_16x16x4_f32(
        false, a, false, b, (short)0, cz, false, false);   // D = areaI+areaJ
    unsigned bals[8];
#pragma unroll
    for (int v = 0; v < 8; ++v) {
      float4 rb = rbox[v + 8*hi];
      float xx1 = fmaxf(rb.x, cb.x), yy1 = fmaxf(rb.y, cb.y);
      float xx2 = fminf(rb.z, cb.z), yy2 = fminf(rb.w, cb.w);
      float iw = fmaxf(xx2 - xx1 + 1.0f, 0.0f);
      float ih = fmaxf(yy2 - yy1 + 1.0f, 0.0f);
      float inter = iw * ih;
      float uni = d[v] - inter;          // legacy IoU denominator
      int row = rowbase + v + 8*hi;
      bool bit = (inter > NMS_TH * uni) && (col > row);
      bals[v] = (unsigned)__builtin_amdgcn_ballot_w32(bit);
    }
    if (l < 16) {                        // lane t owns tile-row t
      unsigned bb = bals[l & 7];
      unsigned bits = (l < 8) ? (bb & 0xFFFFu) : (bb >> 16);
      rw[l][ct >> 1] |= bits << (16 * (ct & 1));
    }
  }
  __syncthreads();
  for (int i = l; i < 16*32; i += 32) {
    int t = i >> 5, w = i & 31;
    mask[(size_t)(rowbase + t)*32 + w] = rw[t][w];
  }
}

// ---------------------------------------------------------------- kernel 4
// greedy scan over bitmask. one wave per class; lane = 32-column word.
__global__ void k_nms_scan(float* __restrict__ ws) {
  const int c = blockIdx.x;
  const int l = threadIdx.x;             // 0..31
  const float* sS = ws + OFF_SS + c*NN;
  const unsigned* mask = (const unsigned*)(ws + OFF_MASK) + (size_t)c*NN*32;
  unsigned valid = 0u;
  for (int b = 0; b < 32; ++b)
    valid |= (sS[l*32 + b] > SCORE_TH ? 1u : 0u) << b;
  unsigned supp = 0u;
  for (int i = 0; i < NN; ++i) {
    int wi = i >> 5, bi = i & 31;
    unsigned kb = ((valid >> bi) & 1u) & (~(supp >> bi) & 1u);
    unsigned kib = (unsigned)__builtin_amdgcn_readlane((int)kb, wi);
    unsigned m = mask[(size_t)i*32 + l];
    supp |= m & (0u - kib);
  }
  unsigned keep = valid & ~supp;
  float* rank = ws + OFF_RANK + c*NN;
  for (int b = 0; b < 32; ++b)
    rank[l*32 + b] = ((keep >> b) & 1u) ? sS[l*32 + b] : -1.0f;
}

// ---------------------------------------------------------------- kernel 5
// top-100 via 100 block-wide argmax passes.
__global__ void k_topk(float* __restrict__ ws, float* __restrict__ out) {
  __shared__ float rv[256];
  __shared__ int   ri[256];
  const int tid = threadIdx.x;
  float* rank = ws + OFF_RANK;
  const float* sB = ws + OFF_SB;
  float* ob = out;             // [100][4]
  float* os = out + DET*4;     // [100]
  int*   ol = (int*)(out + DET*5);   // [100] int32 labels
  for (int p = 0; p < DET; ++p) {
    float best = -1e30f; int bi = 0;
    for (int j = tid; j < NCF*NN; j += 256) {
      float v = rank[j];
      if (v > best || (v == best && j < bi)) { best = v; bi = j; }
    }
    rv[tid] = best; ri[tid] = bi;
    __syncthreads();
    for (int s = 128; s > 0; s >>= 1) {
      if (tid < s) {
        float ov = rv[tid+s]; int oi = ri[tid+s];
        if (ov > rv[tid] || (ov == rv[tid] && oi < ri[tid])) { rv[tid] = ov; ri[tid] = oi; }
      }
      __syncthreads();
    }
    if (tid == 0) {
      float s0 = rv[0]; int m = ri[0];
      int ok = (s0 > 0.0f) ? 1 : 0;
      const float* b4 = sB + (size_t)m*4;
      ob[p*4+0] = ok ? b4[0] : 0.0f;
      ob[p*4+1] = ok ? b4[1] : 0.0f;
      ob[p*4+2] = ok ? b4[2] : 0.0f;
      ob[p*4+3] = ok ? b4[3] : 0.0f;
      os[p] = ok ? s0 : 0.0f;
      ol[p] = ok ? (m / NN + 1) : 0;
      rank[m] = -2.0f;                  // remove winner for next pass
    }
    __syncthreads();                    // also publishes rank[m] block-wide
  }
}

// ---------------------------------------------------------------- launch
extern "C" void kernel_launch(void* const* d_in, const int* in_sizes, int n_in,
                              void* d_out, int out_size, void* d_ws, size_t ws_size,
                              hipStream_t stream) {
  const float* logits = (const float*)d_in[0];   // [1024,81]
  const float* rel    = (const float*)d_in[1];   // [1024,324]
  const float* props  = (const float*)d_in[2];   // [1024,4]
  float* ws  = (float*)d_ws;
  float* out = (float*)d_out;

  k_softmax    <<<NN/256, 256, 0, stream>>>(logits, ws);
  k_sort_decode<<<NCF,    256, 0, stream>>>(rel, props, ws);
  k_mask       <<<dim3(NN/16, NCF), 32, 0, stream>>>(ws);
  k_nms_scan   <<<NCF,     32, 0, stream>>>(ws);
  k_topk       <<<1,      256, 0, stream>>>(ws, out);
}